// InvariantPointAttention_1915555414551
// MI455X (gfx1250) — compile-verified
//
#include <hip/hip_runtime.h>
#include <hip/hip_bf16.h>
#include <math.h>

// ---------------- constants (from reference) ----------------
constexpr int N_RES = 768;
constexpr int HEADS = 12;
constexpr int SKD_  = 16;
constexpr int SVD_  = 16;
constexpr int PKD_  = 4;
constexpr int PVD_  = 8;
constexpr int DIM_  = 384;
constexpr int PD_   = 128;
constexpr int QKW   = HEADS * SKD_;      // 192
constexpr int PQW   = HEADS * PKD_ * 3;  // 144
constexpr int PVW   = HEADS * PVD_ * 3;  // 288
constexpr int FEAT  = HEADS * (SVD_ + PVD_ * 3 + PVD_ + PD_); // 2112

constexpr float SCALAR_SCALE = 0.14433756729740643f;   // (3*16)^-0.5
constexpr float POINT_SCALE  = 0.13608276348795434f;   // (3*4*4.5)^-0.5
constexpr float PAIR_SCALE   = 0.57735026918962576f;   // 3^-0.5
constexpr float EPS_N        = 1e-8f;

typedef __attribute__((ext_vector_type(2))) float v2f;
typedef __attribute__((ext_vector_type(8))) float v8f;

// fp32 WMMA: D(16x16) = A(16x4) * B(4x16) + C  — CDNA5 V_WMMA_F32_16X16X4_F32
__device__ __forceinline__ v8f wmma4(v2f a, v2f b, v8f c) {
  return __builtin_amdgcn_wmma_f32_16x16x4_f32(
      false, a, false, b, (short)0, c, false, false);
}

// ---------------- generic fp32 WMMA GEMM: C[MxN] = A[MxK] @ B[KxN] (+bias) ----
// one wave per 16x16 tile; M%16==0, N%16==0, K%4==0
__global__ void gemm_f32_kernel(const float* __restrict__ A,
                                const float* __restrict__ B,
                                float* __restrict__ C,
                                const float* __restrict__ bias,
                                int M, int N, int K) {
  const int lane = threadIdx.x & 31;
  const int wave = threadIdx.x >> 5;
  const int tile = blockIdx.x * 4 + wave;
  const int tn_cnt = N >> 4;
  if (tile >= (M >> 4) * tn_cnt) return;            // wave-uniform
  const int tm = tile / tn_cnt, tn = tile % tn_cnt;
  const int m = lane & 15, hi = lane >> 4;
  const float* arow = A + (size_t)(tm * 16 + m) * K;
  v8f acc = {};
  for (int k0 = 0; k0 < K; k0 += 4) {
    const int klo = k0 + 2 * hi;
    v2f a = *(const v2f*)(arow + klo);
    v2f b;
    b.x = B[(size_t)(klo + 0) * N + tn * 16 + m];
    b.y = B[(size_t)(klo + 1) * N + tn * 16 + m];
    acc = wmma4(a, b, acc);
  }
  const int ocol = tn * 16 + m;
  const float bb = bias ? bias[ocol] : 0.f;
  for (int r = 0; r < 8; ++r) {
    const int orow = tm * 16 + r + 8 * hi;
    C[(size_t)orow * N + ocol] = acc[r] + bb;
  }
}

// ---------------- rigid transform to global frame + point norms ----------------
__global__ void to_global_kernel(const float* __restrict__ Qp,
                                 const float* __restrict__ Kp,
                                 const float* __restrict__ Vp,
                                 const float* __restrict__ rot,
                                 const float* __restrict__ trans,
                                 float* __restrict__ Qpg, float* __restrict__ Kpg,
                                 float* __restrict__ Vpg,
                                 float* __restrict__ qn, float* __restrict__ kn) {
  const int t = blockIdx.x * blockDim.x + threadIdx.x;
  if (t >= HEADS * N_RES) return;
  const int h = t / N_RES, i = t % N_RES;
  float R[9], T[3];
#pragma unroll
  for (int a = 0; a < 9; ++a) R[a] = rot[i * 9 + a];
#pragma unroll
  for (int c = 0; c < 3; ++c) T[c] = trans[i * 3 + c];

  float qs = 0.f, ks = 0.f;
#pragma unroll
  for (int d = 0; d < PKD_; ++d) {
    float q0 = Qp[(size_t)i * PQW + h * 12 + d * 3 + 0];
    float q1 = Qp[(size_t)i * PQW + h * 12 + d * 3 + 1];
    float q2 = Qp[(size_t)i * PQW + h * 12 + d * 3 + 2];
    float k0 = Kp[(size_t)i * PQW + h * 12 + d * 3 + 0];
    float k1 = Kp[(size_t)i * PQW + h * 12 + d * 3 + 1];
    float k2 = Kp[(size_t)i * PQW + h * 12 + d * 3 + 2];
#pragma unroll
    for (int c = 0; c < 3; ++c) {
      float g = R[c * 3 + 0] * q0 + R[c * 3 + 1] * q1 + R[c * 3 + 2] * q2 + T[c];
      Qpg[(size_t)t * 12 + d * 3 + c] = g;  qs += g * g;
      float gk = R[c * 3 + 0] * k0 + R[c * 3 + 1] * k1 + R[c * 3 + 2] * k2 + T[c];
      Kpg[(size_t)t * 12 + d * 3 + c] = gk; ks += gk * gk;
    }
  }
  qn[t] = qs;  kn[t] = ks;
#pragma unroll
  for (int d = 0; d < PVD_; ++d) {
    float p0 = Vp[(size_t)i * PVW + h * 24 + d * 3 + 0];
    float p1 = Vp[(size_t)i * PVW + h * 24 + d * 3 + 1];
    float p2 = Vp[(size_t)i * PVW + h * 24 + d * 3 + 2];
#pragma unroll
    for (int c = 0; c < 3; ++c)
      Vpg[(size_t)t * 24 + d * 3 + c] =
          R[c * 3 + 0] * p0 + R[c * 3 + 1] * p1 + R[c * 3 + 2] * p2 + T[c];
  }
}

// ---------------- pair bias: L[h,i,j] = PAIR_SCALE*(pair[i,j,:]@Wpb[:,h]+bpb[h]) ----
// Streams 302MB of pairwise with NT loads. One wave per (i, 16-wide j tile).
__global__ void pair_bias_kernel(const float* __restrict__ pair,
                                 const float* __restrict__ Wpb,
                                 const float* __restrict__ bpb,
                                 float* __restrict__ L) {
  const int lane = threadIdx.x & 31, wave = threadIdx.x >> 5;
  const int tile = blockIdx.x * 4 + wave;
  if (tile >= N_RES * (N_RES / 16)) return;          // wave-uniform
  const int i = tile / (N_RES / 16), jt = tile % (N_RES / 16);
  const int m = lane & 15, hi = lane >> 4;
  const float* prow = pair + ((size_t)i * N_RES + jt * 16 + m) * PD_;
  v8f acc = {};
  for (int k0 = 0; k0 < PD_; k0 += 4) {
    const int klo = k0 + 2 * hi;
    v2f a = __builtin_nontemporal_load((const v2f*)(prow + klo));
    v2f b;
    b.x = (m < HEADS) ? Wpb[(klo + 0) * HEADS + m] : 0.f;
    b.y = (m < HEADS) ? Wpb[(klo + 1) * HEADS + m] : 0.f;
    acc = wmma4(a, b, acc);
  }
  const int h = m;                      // D col = head
  if (h < HEADS) {
    const float bb = bpb[h];
#pragma unroll
    for (int r = 0; r < 8; ++r) {
      const int j = jt * 16 + r + 8 * hi;   // D row = j within tile
      L[((size_t)h * N_RES + i) * N_RES + j] = PAIR_SCALE * (acc[r] + bb);
    }
  }
}

// ---------------- scalar QK + point-distance logits (RMW on L) ----------------
__global__ void logits_sp_kernel(const float* __restrict__ Q,
                                 const float* __restrict__ Kmat,
                                 const float* __restrict__ Qpg,
                                 const float* __restrict__ Kpg,
                                 const float* __restrict__ qn,
                                 const float* __restrict__ kn,
                                 const float* __restrict__ pwts,
                                 float* __restrict__ L) {
  const int lane = threadIdx.x & 31, wave = threadIdx.x >> 5;
  const int tile = blockIdx.x * 4 + wave;
  if (tile >= HEADS * 48 * 48) return;               // wave-uniform
  const int h = tile / (48 * 48);
  const int rem = tile % (48 * 48);
  const int it = rem / 48, jt = rem % 48;
  const int m = lane & 15, hi = lane >> 4;

  v8f accS = {}, accP = {};
  const float* qrow = Q    + (size_t)(it * 16 + m) * QKW + h * SKD_;
  const float* krow = Kmat + (size_t)(jt * 16 + m) * QKW + h * SKD_;
#pragma unroll
  for (int k0 = 0; k0 < SKD_; k0 += 4) {
    v2f a = *(const v2f*)(qrow + k0 + 2 * hi);
    v2f b = *(const v2f*)(krow + k0 + 2 * hi);
    accS = wmma4(a, b, accS);
  }
  const float* qprow = Qpg + ((size_t)h * N_RES + it * 16 + m) * 12;
  const float* kprow = Kpg + ((size_t)h * N_RES + jt * 16 + m) * 12;
#pragma unroll
  for (int k0 = 0; k0 < 12; k0 += 4) {
    v2f a = *(const v2f*)(qprow + k0 + 2 * hi);
    v2f b = *(const v2f*)(kprow + k0 + 2 * hi);
    accP = wmma4(a, b, accP);
  }
  const float w = pwts[h];
  const float pw = log1pf(expf(w));                  // softplus
  const float coef = -0.5f * POINT_SCALE * pw;
#pragma unroll
  for (int r = 0; r < 8; ++r) {
    const int i = it * 16 + r + 8 * hi;
    const int j = jt * 16 + m;
    const size_t idx = ((size_t)h * N_RES + i) * N_RES + j;
    const float dist = qn[h * N_RES + i] + kn[h * N_RES + j] - 2.f * accP[r];
    L[idx] += SCALAR_SCALE * accS[r] + coef * dist;
  }
}

// ---------------- softmax over j (one block per (h,i) row) ----------------
__global__ void softmax_kernel(float* __restrict__ L) {
  __shared__ float red[256];
  float* p = L + (size_t)blockIdx.x * N_RES;
  const int tid = threadIdx.x;
  float v[3], mx = -3.4e38f;
#pragma unroll
  for (int r = 0; r < 3; ++r) { v[r] = p[tid + r * 256]; mx = fmaxf(mx, v[r]); }
  red[tid] = mx; __syncthreads();
  for (int s = 128; s > 0; s >>= 1) { if (tid < s) red[tid] = fmaxf(red[tid], red[tid + s]); __syncthreads(); }
  mx = red[0]; __syncthreads();
  float sum = 0.f;
#pragma unroll
  for (int r = 0; r < 3; ++r) { v[r] = expf(v[r] - mx); sum += v[r]; }
  red[tid] = sum; __syncthreads();
  for (int s = 128; s > 0; s >>= 1) { if (tid < s) red[tid] += red[tid + s]; __syncthreads(); }
  const float inv = 1.f / red[0];
#pragma unroll
  for (int r = 0; r < 3; ++r) p[tid + r * 256] = v[r] * inv;
}

// ---------------- attn @ V and attn @ vp_g (K=768 split over 4 waves) ----------
__global__ void attn_v_kernel(const float* __restrict__ attn,
                              const float* __restrict__ V,
                              const float* __restrict__ Vpg,
                              float* __restrict__ feats,
                              float* __restrict__ ptsg) {
  __shared__ float sS[4 * 256], sP0[4 * 256], sP1[4 * 256];
  const int lane = threadIdx.x & 31, wave = threadIdx.x >> 5;
  const int h = blockIdx.x / 48, it = blockIdx.x % 48;
  const int m = lane & 15, hi = lane >> 4;
  v8f aS = {}, aP0 = {}, aP1 = {};
  const float* arow = attn + ((size_t)h * N_RES + it * 16 + m) * N_RES;
  const int kbeg = wave * 192, kend = kbeg + 192;
  for (int k0 = kbeg; k0 < kend; k0 += 4) {
    const int klo = k0 + 2 * hi;
    v2f a = *(const v2f*)(arow + klo);
    v2f b;
    b.x = V[(size_t)(klo + 0) * QKW + h * SVD_ + m];
    b.y = V[(size_t)(klo + 1) * QKW + h * SVD_ + m];
    aS = wmma4(a, b, aS);
    const float* vp = Vpg + ((size_t)h * N_RES + klo) * 24;
    v2f c0; c0.x = vp[m];       c0.y = vp[24 + m];
    aP0 = wmma4(a, c0, aP0);
    v2f c1;
    c1.x = (m < 8) ? vp[16 + m]      : 0.f;
    c1.y = (m < 8) ? vp[24 + 16 + m] : 0.f;
    aP1 = wmma4(a, c1, aP1);
  }
#pragma unroll
  for (int r = 0; r < 8; ++r) {
    const int e = (r + 8 * hi) * 16 + m;
    sS[wave * 256 + e] = aS[r]; sP0[wave * 256 + e] = aP0[r]; sP1[wave * 256 + e] = aP1[r];
  }
  __syncthreads();
  for (int e = threadIdx.x; e < 256; e += 128) {
    const float s  = sS [e] + sS [256 + e] + sS [512 + e] + sS [768 + e];
    const float p0 = sP0[e] + sP0[256 + e] + sP0[512 + e] + sP0[768 + e];
    const float p1 = sP1[e] + sP1[256 + e] + sP1[512 + e] + sP1[768 + e];
    const int mm = e >> 4, nn = e & 15;
    const int i = it * 16 + mm;
    feats[(size_t)i * FEAT + h * SVD_ + nn] = s;          // res_scalar block
    float* pg = ptsg + ((size_t)h * N_RES + i) * 24;
    pg[nn] = p0;
    if (nn < 8) pg[16 + nn] = p1;
  }
}

// ---------------- inverse rigid transform + norms → feats ----------------
__global__ void pts_finalize_kernel(const float* __restrict__ ptsg,
                                    const float* __restrict__ rot,
                                    const float* __restrict__ trans,
                                    float* __restrict__ feats) {
  const int t = blockIdx.x * blockDim.x + threadIdx.x;
  if (t >= HEADS * N_RES) return;
  const int h = t / N_RES, i = t % N_RES;
  float R[9], T[3];
#pragma unroll
  for (int a = 0; a < 9; ++a) R[a] = rot[i * 9 + a];
#pragma unroll
  for (int c = 0; c < 3; ++c) T[c] = trans[i * 3 + c];
  const float* pg = ptsg + (size_t)t * 24;
#pragma unroll
  for (int d = 0; d < PVD_; ++d) {
    const float g0 = pg[d * 3 + 0] - T[0];
    const float g1 = pg[d * 3 + 1] - T[1];
    const float g2 = pg[d * 3 + 2] - T[2];
    const float o0 = R[0] * g0 + R[3] * g1 + R[6] * g2;   // R^T
    const float o1 = R[1] * g0 + R[4] * g1 + R[7] * g2;
    const float o2 = R[2] * g0 + R[5] * g1 + R[8] * g2;
    float* f = feats + (size_t)i * FEAT;
    f[192 + h * 24 + d * 3 + 0] = o0;
    f[192 + h * 24 + d * 3 + 1] = o1;
    f[192 + h * 24 + d * 3 + 2] = o2;
    f[480 + h * PVD_ + d] = sqrtf(o0 * o0 + o1 * o1 + o2 * o2 + EPS_N);
  }
}

// ---------------- res_pair: feats[i, h, d] = Σ_j attn[h,i,j]*pair[i,j,d] ----------
// Second 302MB streaming pass (NT loads). One wave per (i, 16-wide d tile), M=heads.
__global__ void res_pair_kernel(const float* __restrict__ attn,
                                const float* __restrict__ pair,
                                float* __restrict__ feats) {
  const int lane = threadIdx.x & 31, wave = threadIdx.x >> 5;
  const int tile = blockIdx.x * 4 + wave;
  if (tile >= N_RES * (PD_ / 16)) return;            // wave-uniform
  const int i = tile >> 3, dt = tile & 7;
  const int m = lane & 15, hi = lane >> 4;
  const float maskA = (m < HEADS) ? 1.f : 0.f;
  const float* arow = attn + ((size_t)(m < HEADS ? m : 0) * N_RES + i) * N_RES;
  const float* pcol = pair + (size_t)i * N_RES * PD_ + dt * 16 + m;
  v8f acc = {};
  for (int k0 = 0; k0 < N_RES; k0 += 4) {
    const int klo = k0 + 2 * hi;
    v2f a = *(const v2f*)(arow + klo);
    a.x *= maskA; a.y *= maskA;
    v2f b;
    b.x = __builtin_nontemporal_load(pcol + (size_t)(klo + 0) * PD_);
    b.y = __builtin_nontemporal_load(pcol + (size_t)(klo + 1) * PD_);
    acc = wmma4(a, b, acc);
  }
#pragma unroll
  for (int r = 0; r < 8; ++r) {
    const int h = r + 8 * hi;                         // D row = head
    if (h < HEADS)
      feats[(size_t)i * FEAT + 576 + h * PD_ + dt * 16 + m] = acc[r];
  }
}

// ---------------- host ----------------
extern "C" void kernel_launch(void* const* d_in, const int* in_sizes, int n_in,
                              void* d_out, int out_size, void* d_ws, size_t ws_size,
                              hipStream_t stream) {
  const float* x     = (const float*)d_in[0];
  const float* pair  = (const float*)d_in[1];
  const float* rot   = (const float*)d_in[2];
  const float* trans = (const float*)d_in[3];
  const float* Wq    = (const float*)d_in[4];
  const float* Wk    = (const float*)d_in[5];
  const float* Wv    = (const float*)d_in[6];
  const float* Wpq   = (const float*)d_in[7];
  const float* Wpk   = (const float*)d_in[8];
  const float* Wpv   = (const float*)d_in[9];
  const float* Wpb   = (const float*)d_in[10];
  const float* bpb   = (const float*)d_in[11];
  const float* pwts  = (const float*)d_in[12];
  const float* Wout  = (const float*)d_in[13];
  const float* bout  = (const float*)d_in[14];

  float* ws = (float*)d_ws;
  size_t off = 0;
  auto alloc = [&](size_t n) { float* p = ws + off; off += n; return p; };
  float* Q    = alloc((size_t)N_RES * QKW);
  float* Km   = alloc((size_t)N_RES * QKW);
  float* V    = alloc((size_t)N_RES * QKW);
  float* Qp   = alloc((size_t)N_RES * PQW);
  float* Kp   = alloc((size_t)N_RES * PQW);
  float* Vp   = alloc((size_t)N_RES * PVW);
  float* Qpg  = alloc((size_t)HEADS * N_RES * 12);
  float* Kpg  = alloc((size_t)HEADS * N_RES * 12);
  float* Vpg  = alloc((size_t)HEADS * N_RES * 24);
  float* qn   = alloc((size_t)HEADS * N_RES);
  float* kn   = alloc((size_t)HEADS * N_RES);
  float* L    = alloc((size_t)HEADS * N_RES * N_RES);   // logits -> attn (28.3MB)
  float* feats= alloc((size_t)N_RES * FEAT);
  float* ptsg = alloc((size_t)HEADS * N_RES * 24);
  (void)ws_size; (void)in_sizes; (void)n_in; (void)out_size;

  auto gemm = [&](const float* A, const float* B, float* C, const float* bias,
                  int M, int N, int K) {
    int tiles = (M / 16) * (N / 16);
    gemm_f32_kernel<<<(tiles + 3) / 4, 128, 0, stream>>>(A, B, C, bias, M, N, K);
  };

  // projections
  gemm(x, Wq,  Q,  nullptr, N_RES, QKW, DIM_);
  gemm(x, Wk,  Km, nullptr, N_RES, QKW, DIM_);
  gemm(x, Wv,  V,  nullptr, N_RES, QKW, DIM_);
  gemm(x, Wpq, Qp, nullptr, N_RES, PQW, DIM_);
  gemm(x, Wpk, Kp, nullptr, N_RES, PQW, DIM_);
  gemm(x, Wpv, Vp, nullptr, N_RES, PVW, DIM_);

  to_global_kernel<<<(HEADS * N_RES + 255) / 256, 256, 0, stream>>>(
      Qp, Kp, Vp, rot, trans, Qpg, Kpg, Vpg, qn, kn);

  pair_bias_kernel<<<(N_RES * 48) / 4, 128, 0, stream>>>(pair, Wpb, bpb, L);
  logits_sp_kernel<<<(HEADS * 48 * 48) / 4, 128, 0, stream>>>(
      Q, Km, Qpg, Kpg, qn, kn, pwts, L);
  softmax_kernel<<<HEADS * N_RES, 256, 0, stream>>>(L);

  attn_v_kernel<<<HEADS * 48, 128, 0, stream>>>(L, V, Vpg, feats, ptsg);
  pts_finalize_kernel<<<(HEADS * N_RES + 255) / 256, 256, 0, stream>>>(
      ptsg, rot, trans, feats);
  res_pair_kernel<<<(N_RES * 8) / 4, 128, 0, stream>>>(L, pair, feats);

  // final projection with bias
  gemm(feats, Wout, (float*)d_out, bout, N_RES, DIM_, FEAT);
}